// ContourLoss_13915694039310
// MI455X (gfx1250) — compile-verified
//
#include <hip/hip_runtime.h>
#include <math.h>

// Problem constants (match reference)
#define B_SAMPLES 512
#define D_IN      512
#define N_CLASSES 16
#define N_NODES   100000
#define WAVES_PER_BLOCK 8

typedef __attribute__((ext_vector_type(2))) float v2f;
typedef __attribute__((ext_vector_type(8))) float v8f;

// ---------------------------------------------------------------------------
// Kernel 0: zero the scalar output (harness poisons it with 0xAA)
// ---------------------------------------------------------------------------
__global__ void zero_out_kernel(float* out) {
    if (threadIdx.x == 0 && blockIdx.x == 0) out[0] = 0.0f;
}

// ---------------------------------------------------------------------------
// Kernel 1: per-sample prep. One thread per sample:
//   probs = softmax(logits_row), pred_class = argmax(logits_row),
//   emb = train_data_row @ W + b   (512 -> 2 projection)
// ---------------------------------------------------------------------------
__global__ __launch_bounds__(256) void prep_samples_kernel(
    const float* __restrict__ logits,      // [B,16]
    const float* __restrict__ train_data,  // [B,512]
    const float* __restrict__ W,           // [512,2] row-major
    const float* __restrict__ bias,        // [2]
    float* __restrict__ probs,             // [B,16]
    int*   __restrict__ pred_class,        // [B]
    float* __restrict__ emb)               // [B,2]
{
    int b = blockIdx.x * blockDim.x + threadIdx.x;
    if (b >= B_SAMPLES) return;

    float l[N_CLASSES];
    float mx = -INFINITY;
    int   am = 0;
    float amv = -INFINITY;
    #pragma unroll
    for (int c = 0; c < N_CLASSES; ++c) {
        l[c] = logits[b * N_CLASSES + c];
        mx = fmaxf(mx, l[c]);
        bool upd = l[c] > amv;         // strict > : first-occurrence argmax
        amv = upd ? l[c] : amv;
        am  = upd ? c    : am;
    }
    float s = 0.0f;
    #pragma unroll
    for (int c = 0; c < N_CLASSES; ++c) { l[c] = expf(l[c] - mx); s += l[c]; }
    float inv = 1.0f / s;
    #pragma unroll
    for (int c = 0; c < N_CLASSES; ++c) probs[b * N_CLASSES + c] = l[c] * inv;
    pred_class[b] = am;

    float e0 = bias[0], e1 = bias[1];
    const float* td = train_data + (size_t)b * D_IN;
    for (int j = 0; j < D_IN; ++j) {
        float x = td[j];
        e0 = fmaf(x, W[2 * j + 0], e0);
        e1 = fmaf(x, W[2 * j + 1], e1);
    }
    emb[2 * b + 0] = e0;
    emb[2 * b + 1] = e1;
}

// ---------------------------------------------------------------------------
// Kernel 2: per-node prep. One thread per node:
//   node_class = argmax(node_row), inv_nnorm = 1/||node_row||
// (dividing by ||probs|| is a per-row positive scale -> argmax-invariant, dropped)
// ---------------------------------------------------------------------------
__global__ __launch_bounds__(256) void prep_nodes_kernel(
    const float* __restrict__ all_nodes,   // [N,16]
    int*   __restrict__ node_class,        // [N]
    float* __restrict__ inv_nnorm)         // [N]
{
    int n = blockIdx.x * blockDim.x + threadIdx.x;
    if (n >= N_NODES) return;
    const float* row = all_nodes + (size_t)n * N_CLASSES;
    float ss = 0.0f;
    int   am = 0;
    float amv = -INFINITY;
    #pragma unroll
    for (int c = 0; c < N_CLASSES; ++c) {
        float v = row[c];
        ss = fmaf(v, v, ss);
        bool upd = v > amv;
        amv = upd ? v : amv;
        am  = upd ? c : am;
    }
    node_class[n] = am;
    inv_nnorm[n] = 1.0f / fmaxf(sqrtf(ss), 1e-12f);
}

// ---------------------------------------------------------------------------
// Kernel 3: fused masked nearest-neighbor search + MSE accumulation.
// grid.x = 32 (one block per 16-sample stripe), 256 threads = 8 waves.
// Each wave strides over 16-node tiles; each tile = 4 chained
// V_WMMA_F32_16X16X4_F32 ops (K = 16 = 4 x 4).
//
// Fragment layouts (CDNA5 ISA 7.12.2, 32-bit, wave32), lane l, h=l/16, j=l%16:
//   A  VGPR v : A[M=j][K = v + 2h]          -> float2 from probs row j
//   B  VGPR v : B[K = v + 2h][N=j]          -> float2 from all_nodes row j
//   C  VGPR r : sim[M = r + 8h][N = j]
// ---------------------------------------------------------------------------
__global__ __launch_bounds__(256) void nn_search_kernel(
    const float* __restrict__ probs,       // [512,16]
    const int*   __restrict__ pred_class,  // [512]
    const float* __restrict__ all_nodes,   // [100000,16]
    const int*   __restrict__ node_class,  // [100000]
    const float* __restrict__ inv_nnorm,   // [100000]
    const float* __restrict__ emb,         // [512,2]
    const float* __restrict__ nodes2d,     // [100000,2]
    float* __restrict__ out)               // [1]
{
    const int row0 = blockIdx.x * 16;      // first sample of this stripe
    const int tid  = threadIdx.x;
    const int wave = tid >> 5;
    const int lane = tid & 31;
    const int half = lane >> 4;
    const int j    = lane & 15;            // M index for A, N index for B/C

    // A fragments for the 4 K-slices (reused across all node tiles)
    v2f a[4];
    #pragma unroll
    for (int k = 0; k < 4; ++k) {
        const float* p = probs + (size_t)(row0 + j) * N_CLASSES + 4 * k + 2 * half;
        v2f t = { p[0], p[1] };
        a[k] = t;
    }
    // pred_class for the 8 sample rows this lane's C VGPRs map to (m = r + 8h)
    int pc[8];
    #pragma unroll
    for (int r = 0; r < 8; ++r) pc[r] = pred_class[row0 + r + 8 * half];

    float best_val[8];
    int   best_idx[8];
    #pragma unroll
    for (int r = 0; r < 8; ++r) { best_val[r] = -INFINITY; best_idx[r] = 0; }

    const int n_tiles = N_NODES / 16;      // 6250 (exact)
    for (int t = wave; t < n_tiles; t += WAVES_PER_BLOCK) {
        const int n = t * 16 + j;          // this lane's node (column N of tile)
        const int   nc    = node_class[n];
        const float inv_n = inv_nnorm[n];

        v8f c = {};
        #pragma unroll
        for (int k = 0; k < 4; ++k) {
            const float* q = all_nodes + (size_t)n * N_CLASSES + 4 * k + 2 * half;
            v2f bfrag = { q[0], q[1] };
            c = __builtin_amdgcn_wmma_f32_16x16x4_f32(
                    /*neg_a=*/false, a[k], /*neg_b=*/false, bfrag,
                    /*c_mod=*/(short)0, c, /*reuse_a=*/false, /*reuse_b=*/false);
        }

        // masked running argmax (strict > keeps first-occurrence semantics;
        // selects, not branches, so EXEC stays all-ones around the WMMAs)
        #pragma unroll
        for (int r = 0; r < 8; ++r) {
            float v = (pc[r] == nc) ? (c[r] * inv_n) : -INFINITY;
            bool upd = v > best_val[r];
            best_val[r] = upd ? v : best_val[r];
            best_idx[r] = upd ? n : best_idx[r];
        }
    }

    // Reduce over N within each 16-lane half (tie -> smaller node index),
    // then across the 8 waves via LDS.
    __shared__ float s_val[WAVES_PER_BLOCK][16];
    __shared__ int   s_idx[WAVES_PER_BLOCK][16];

    #pragma unroll
    for (int r = 0; r < 8; ++r) {
        float v = best_val[r];
        int   i = best_idx[r];
        #pragma unroll
        for (int off = 1; off < 16; off <<= 1) {
            float ov = __shfl_xor(v, off, 32);
            int   oi = __shfl_xor(i, off, 32);
            bool take = (ov > v) || (ov == v && oi < i);
            v = take ? ov : v;
            i = take ? oi : i;
        }
        if (j == r) {                       // one writer per (r, half)
            s_val[wave][r + 8 * half] = v;
            s_idx[wave][r + 8 * half] = i;
        }
    }
    __syncthreads();

    if (tid < 16) {
        float bv = -INFINITY;
        int   bi = 0;
        #pragma unroll
        for (int w = 0; w < WAVES_PER_BLOCK; ++w) {
            float v = s_val[w][tid];
            int   i = s_idx[w][tid];
            bool take = (v > bv) || (v == bv && i < bi);
            bv = take ? v : bv;
            bi = take ? i : bi;
        }
        const int m = row0 + tid;
        float t0 = nodes2d[2 * bi + 0];
        float t1 = nodes2d[2 * bi + 1];
        float d0 = emb[2 * m + 0] - t0;
        float d1 = emb[2 * m + 1] - t1;
        float per_sample = 0.5f * (d0 * d0 + d1 * d1);   // mean over D_2D=2
        atomicAdd(out, per_sample * (1.0f / (float)B_SAMPLES));
    }
}

// ---------------------------------------------------------------------------
// Launcher
// ---------------------------------------------------------------------------
extern "C" void kernel_launch(void* const* d_in, const int* in_sizes, int n_in,
                              void* d_out, int out_size, void* d_ws, size_t ws_size,
                              hipStream_t stream) {
    const float* logits     = (const float*)d_in[0];   // [512,16]
    const float* train_data = (const float*)d_in[1];   // [512,512]
    const float* all_nodes  = (const float*)d_in[2];   // [100000,16]
    const float* nodes2d    = (const float*)d_in[3];   // [100000,2]
    const float* W          = (const float*)d_in[4];   // [512,2]
    const float* bias       = (const float*)d_in[5];   // [2]
    float* out = (float*)d_out;

    // Workspace layout (256B-aligned chunks), total ~0.85 MB
    char* ws = (char*)d_ws;
    float* probs      = (float*)(ws);                       // 512*16*4  = 32768
    float* emb        = (float*)(ws + 32768);               // 512*2*4   = 4096
    int*   pred_class = (int*)  (ws + 32768 + 4096);        // 512*4     = 2048
    int*   node_class = (int*)  (ws + 40960);               // 100000*4  = 400000
    float* inv_nnorm  = (float*)(ws + 40960 + 401408);      // 100000*4  = 400000
    (void)in_sizes; (void)n_in; (void)out_size; (void)ws_size;

    zero_out_kernel<<<1, 32, 0, stream>>>(out);

    prep_samples_kernel<<<(B_SAMPLES + 255) / 256, 256, 0, stream>>>(
        logits, train_data, W, bias, probs, pred_class, emb);

    prep_nodes_kernel<<<(N_NODES + 255) / 256, 256, 0, stream>>>(
        all_nodes, node_class, inv_nnorm);

    nn_search_kernel<<<B_SAMPLES / 16, 32 * WAVES_PER_BLOCK, 0, stream>>>(
        probs, pred_class, all_nodes, node_class, inv_nnorm, emb, nodes2d, out);
}